// Concat_73607149519362
// MI455X (gfx1250) — compile-verified
//
#include <hip/hip_runtime.h>

// ---------------------------------------------------------------------------
// attn[b,i,j] = (decoder[b,i,:]·w_dec) + (encoder[b,j,:]·w_enc) + bias
// Phase 1: row-dot reductions via V_WMMA_F32_16X16X4_F32 (f32 precision kept)
// Phase 2: streaming broadcast-add of the 64 MB output
// ---------------------------------------------------------------------------

#define DIM       1024
#define BATCH     32
#define DEC_LEN   512
#define ENC_LEN   1024

#define ENC_ROWS  (BATCH * ENC_LEN)   // 32768
#define DEC_ROWS  (BATCH * DEC_LEN)   // 16384
#define ENC_TILES (ENC_ROWS / 16)     // 2048
#define DEC_TILES (DEC_ROWS / 16)     // 1024
#define ALL_TILES (ENC_TILES + DEC_TILES) // 3072

typedef __attribute__((ext_vector_type(2))) float v2f;
typedef __attribute__((ext_vector_type(4))) float v4f;
typedef __attribute__((ext_vector_type(8))) float v8f;

// ---------------------------------------------------------------------------
// Phase 1: one wave32 per 16-row tile.
// A (16x4 f32) ISA layout: lanes 0-15 -> rows, VGPR{0,1} = K{0,1}; lanes 16-31
// hold K{2,3} for the same rows. So lane l loads float2 at
//   states[row*DIM + kb + (l>=16 ? 2 : 0)]
// B (4x16) carries w[kb+k] broadcast across all N columns -> every lane's B
// fragment is the same float2 of w (mirrors the A K-addressing).
// C/D (16x16 f32): lanes 0-15 VGPR v = row v, lanes 16-31 VGPR v = row v+8;
// columns are identical, so lanes 0 and 16 emit the 16 results.
// ---------------------------------------------------------------------------
__global__ __launch_bounds__(256) void proj_wmma_kernel(
    const float* __restrict__ dec_states,
    const float* __restrict__ enc_states,
    const float* __restrict__ w,          // [2*DIM]: w_enc | w_dec
    float* __restrict__ enc_proj,         // [ENC_ROWS]
    float* __restrict__ dec_proj)         // [DEC_ROWS]
{
    const int wave = (blockIdx.x * blockDim.x + threadIdx.x) >> 5;
    const int lane = threadIdx.x & 31;
    if (wave >= ALL_TILES) return;

    const float* src;
    const float* wv;
    float*       dst;
    int          tile16;
    if (wave < ENC_TILES) {
        src    = enc_states;
        wv     = w;               // w_enc
        dst    = enc_proj;
        tile16 = wave * 16;
    } else {
        src    = dec_states;
        wv     = w + DIM;         // w_dec
        dst    = dec_proj;
        tile16 = (wave - ENC_TILES) * 16;
    }

    const int row  = tile16 + (lane & 15);
    const int koff = (lane >> 4) * 2;               // 0 for lanes 0-15, 2 for 16-31
    const float* ap = src + (size_t)row * DIM + koff;
    const float* bp = wv + koff;

    v8f c = {0.f, 0.f, 0.f, 0.f, 0.f, 0.f, 0.f, 0.f};

    for (int kb = 0; kb < DIM; kb += 32) {
        __builtin_prefetch(ap + kb + 128, 0, 1);    // global_prefetch_b8, stream ahead
#pragma unroll
        for (int k2 = 0; k2 < 32; k2 += 4) {
            v2f a = *(const v2f*)(ap + kb + k2);
            v2f b = *(const v2f*)(bp + kb + k2);
            // D = A(16x4) * B(4x16) + C   -> v_wmma_f32_16x16x4_f32
            c = __builtin_amdgcn_wmma_f32_16x16x4_f32(
                    /*neg_a=*/false, a, /*neg_b=*/false, b,
                    /*c_mod=*/(short)0, c, /*reuse_a=*/false, /*reuse_b=*/false);
        }
    }

    // Columns are identical; lane 0 holds rows 0-7, lane 16 holds rows 8-15.
    if ((lane & 15) == 0) {
        float* o = dst + tile16 + (lane >> 4) * 8;
#pragma unroll
        for (int v = 0; v < 8; ++v) o[v] = c[v];
    }
}

// ---------------------------------------------------------------------------
// Phase 2: out[b,i,j] = dec_proj[b,i] + enc_proj[b,j] + bias, float4 stores.
// 32*512*1024 / 4 = 4,194,304 float4 elements.
// ---------------------------------------------------------------------------
__global__ __launch_bounds__(256) void bcast_add_kernel(
    const float* __restrict__ enc_proj,
    const float* __restrict__ dec_proj,
    const float* __restrict__ bias,
    float* __restrict__ out)
{
    const unsigned idx = blockIdx.x * blockDim.x + threadIdx.x; // float4 index
    const unsigned j4 = idx & 255u;          // 256 float4 per j-row of 1024
    const unsigned i  = (idx >> 8) & 511u;   // 512 decoder positions
    const unsigned b  = idx >> 17;           // 32 batches

    const float d = dec_proj[b * DEC_LEN + i] + bias[0];
    const v4f  e = *(const v4f*)(enc_proj + b * ENC_LEN + j4 * 4);
    v4f r = {e.x + d, e.y + d, e.z + d, e.w + d};
    *((v4f*)out + idx) = r;
}

extern "C" void kernel_launch(void* const* d_in, const int* in_sizes, int n_in,
                              void* d_out, int out_size, void* d_ws, size_t ws_size,
                              hipStream_t stream) {
    const float* dec_states = (const float*)d_in[0]; // (32, 512, 1024)
    const float* enc_states = (const float*)d_in[1]; // (32, 1024, 1024)
    // d_in[2] = step (int scalar), unused
    const float* mlp_weight = (const float*)d_in[3]; // (1, 2048)
    const float* mlp_bias   = (const float*)d_in[4]; // (1,)

    float* enc_proj = (float*)d_ws;                  // 32768 f32
    float* dec_proj = enc_proj + ENC_ROWS;           // 16384 f32

    // Phase 1: 3072 wave-tiles, 8 waves per 256-thread block -> 384 blocks.
    proj_wmma_kernel<<<ALL_TILES / 8, 256, 0, stream>>>(
        dec_states, enc_states, mlp_weight, enc_proj, dec_proj);

    // Phase 2: 4,194,304 float4 threads -> 16384 blocks of 256.
    bcast_add_kernel<<<(BATCH * DEC_LEN * ENC_LEN / 4) / 256, 256, 0, stream>>>(
        enc_proj, dec_proj, mlp_bias, (float*)d_out);
}